// Contour_to_mask_77549929497022
// MI455X (gfx1250) — compile-verified
//
#include <hip/hip_runtime.h>
#include <math.h>

typedef __attribute__((ext_vector_type(2))) float v2f;
typedef __attribute__((ext_vector_type(8))) float v8f;

#define KV 64

__global__ __launch_bounds__(256) void contour_mask_wmma(
    const float* __restrict__ contour, float* __restrict__ out)
{
  __shared__ float sc[KV * 2];

  const int tid  = threadIdx.x;
  const int wave = tid >> 5;          // 0..7
  const int lane = tid & 31;
  const int cidx  = blockIdx.x >> 9;  // contour 0..7
  const int chunk = blockIdx.x & 511; // 128-pixel chunk

  __builtin_prefetch(contour + (size_t)cidx * KV * 2, 0, 1); // global_prefetch_b8

  if (tid < KV * 2) sc[tid] = contour[cidx * KV * 2 + tid];
  __syncthreads();

  const int pbase = chunk * 128 + wave * 16;  // first pixel of this wave's tile
  const int ii    = pbase >> 8;               // mesh row (constant across tile)
  const int jj0   = pbase & 255;              // mesh col base
  const int mcol  = lane & 15;                // tile-local pixel for A / vertex for B,C
  const bool hi   = lane >= 16;

  const float inv = 1.0f / 256.0f;
  const float mx  = (float)ii * inv;
  const float myc = (float)(jj0 + mcol) * inv;
  const float q   = mx * mx + myc * myc;

  // A (16x4 f32): lanes 0-15 -> K=0,1 ; lanes 16-31 -> K=2,3 ; M = lane%16
  v2f A;
  A.x = hi ? myc : 1.0f;  // K=2 : K=0
  A.y = hi ? q   : mx;    // K=3 : K=1

  float acc[8];
  #pragma unroll
  for (int r = 0; r < 8; ++r) acc[r] = 0.0f;

  const float eps = 1e-5f;

  #pragma unroll
  for (int g = 0; g < 4; ++g) {
    const int n  = g * 16 + mcol;
    const int n1 = (n + 1) & (KV - 1);
    const float cnx = sc[2 * n],  cny = sc[2 * n + 1];
    const float c1x = sc[2 * n1], c1y = sc[2 * n1 + 1];

    // B (4x16 f32) coefficient columns for vertex n (N = lane%16)
    v2f Bna, Bnb, Bdt, Bcr;
    Bna.x = hi ? (-2.0f * cny)     : (cnx * cnx + cny * cny);
    Bna.y = hi ? 1.0f              : (-2.0f * cnx);
    Bnb.x = hi ? (-2.0f * c1y)     : (c1x * c1x + c1y * c1y);
    Bnb.y = hi ? 1.0f              : (-2.0f * c1x);
    Bdt.x = hi ? (-(cny + c1y))    : (cnx * c1x + cny * c1y);
    Bdt.y = hi ? 1.0f              : (-(cnx + c1x));
    Bcr.x = hi ? (cnx - c1x)       : (cny * c1x - cnx * c1y);
    Bcr.y = hi ? 0.0f              : (c1y - cny);

    v8f z = {};
    v8f Cna = __builtin_amdgcn_wmma_f32_16x16x4_f32(false, A, false, Bna, (short)0, z, false, false);
    v8f Cnb = __builtin_amdgcn_wmma_f32_16x16x4_f32(false, A, false, Bnb, (short)0, z, false, false);
    v8f Cdt = __builtin_amdgcn_wmma_f32_16x16x4_f32(false, A, false, Bdt, (short)0, z, false, false);
    v8f Ccr = __builtin_amdgcn_wmma_f32_16x16x4_f32(false, A, false, Bcr, (short)0, z, false, false);

    #pragma unroll
    for (int r = 0; r < 8; ++r) {
      float na2 = Cna[r], nb2 = Cnb[r], dt = Cdt[r], cr = Ccr[r];
      // exact recompute near a vertex, where the expanded form cancels
      if (na2 < 1e-2f || nb2 < 1e-2f) {
        const int m = r + (hi ? 8 : 0);
        const float myr = (float)(jj0 + m) * inv;
        const float ax = cnx - mx, ay = cny - myr;
        const float bx = c1x - mx, by = c1y - myr;
        na2 = ax * ax + ay * ay;
        nb2 = bx * bx + by * by;
        dt  = ax * bx + ay * by;
        cr  = ay * bx - ax * by;
      }
      float denom = sqrtf(na2) * sqrtf(nb2);
      float ratio = dt / denom;
      ratio = fminf(fmaxf(ratio, -1.0f + eps), 1.0f - eps);
      float ang = acosf(ratio);
      float sg  = tanhf(100000.0f * cr);
      acc[r] += sg * ang;
    }
  }

  // reduce over 16 vertices-per-group lanes (stays within each 16-lane half)
  #pragma unroll
  for (int r = 0; r < 8; ++r) {
    float v = acc[r];
    v += __shfl_xor(v, 1, 32);
    v += __shfl_xor(v, 2, 32);
    v += __shfl_xor(v, 4, 32);
    v += __shfl_xor(v, 8, 32);
    acc[r] = v;
  }

  if (mcol == 0) {
    const float inv2pi = 0.15915494309189535f;
    const int mb = hi ? 8 : 0;
    #pragma unroll
    for (int r = 0; r < 8; ++r) {
      float v = fminf(fabsf(acc[r]) * inv2pi, 1.0f);
      out[(size_t)cidx * 65536 + pbase + mb + r] = v;
    }
  }
}

extern "C" void kernel_launch(void* const* d_in, const int* in_sizes, int n_in,
                              void* d_out, int out_size, void* d_ws, size_t ws_size,
                              hipStream_t stream) {
  (void)in_sizes; (void)n_in; (void)out_size; (void)d_ws; (void)ws_size;
  const float* contour = (const float*)d_in[0];
  float* out = (float*)d_out;
  contour_mask_wmma<<<dim3(4096), dim3(256), 0, stream>>>(contour, out);
}